// GELU207_39857296507303
// MI455X (gfx1250) — compile-verified
//
#include <hip/hip_runtime.h>
#include <stdint.h>
#include <math.h>

// ---------------------------------------------------------------------------
// Fused GELU + z-score top/bot-k gating + cosine gate  (MI455X / gfx1250)
//
// Memory-bound op (512 MB traffic @ 23.3 TB/s ~= 22us floor, AI ~2 FLOP/B):
// no matmul -> WMMA is not applicable; the CDNA5 path that matters is the
// async global->LDS copy (ASYNCcnt) to stage each 16KB row while per-column
// stats load, plus wave32 shuffle reductions for the row dot-products and
// the 32 argmax/argmin rounds of the k=16 selections.
// ---------------------------------------------------------------------------

#define DD     4096
#define BLK    256
#define NSLOT  (DD / BLK)   // 16 columns per thread, stride-256 layout
#define KSEL   16
#define WAVES  (BLK / 32)   // 8 waves (wave32)

typedef int v4i_a __attribute__((ext_vector_type(4)));
typedef __attribute__((address_space(1))) v4i_a* gptr_v4i;  // global
typedef __attribute__((address_space(3))) v4i_a* lptr_v4i;  // LDS

// ---- CDNA5 async global->LDS copy (tracked with ASYNCcnt) ------------------
__device__ __forceinline__ void async_b128_to_lds(const float* g, float* l) {
#if __has_builtin(__builtin_amdgcn_global_load_async_to_lds_b128)
  float* gm = const_cast<float*>(g);
  __builtin_amdgcn_global_load_async_to_lds_b128(
      (gptr_v4i)gm, (lptr_v4i)l, 0, 0);
#else
  // GV mode: vdst = LDS byte address, vaddr = 64-bit global address
  unsigned loff = (unsigned)(uintptr_t)l;  // flat LDS addr low 32 bits == LDS offset
  asm volatile("global_load_async_to_lds_b128 %0, %1, off"
               :: "v"(loff), "v"(g) : "memory");
#endif
}

__device__ __forceinline__ void wait_async0() {
#if __has_builtin(__builtin_amdgcn_s_wait_asynccnt)
  __builtin_amdgcn_s_wait_asynccnt(0);
#else
  asm volatile("s_wait_asynccnt 0x0" ::: "memory");
#endif
}

// ---- wave32 shuffle helpers ------------------------------------------------
__device__ __forceinline__ unsigned long long shfl_xor_u64(unsigned long long v, int m) {
  unsigned lo = (unsigned)v, hi = (unsigned)(v >> 32);
  lo = __shfl_xor(lo, m, 32);
  hi = __shfl_xor(hi, m, 32);
  return ((unsigned long long)hi << 32) | lo;
}

__device__ __forceinline__ unsigned long long wave_umax(unsigned long long v) {
#pragma unroll
  for (int o = 16; o > 0; o >>= 1) {
    unsigned long long w = shfl_xor_u64(v, o);
    v = (w > v) ? w : v;
  }
  return v;
}

__device__ __forceinline__ unsigned long long wave_umin(unsigned long long v) {
#pragma unroll
  for (int o = 16; o > 0; o >>= 1) {
    unsigned long long w = shfl_xor_u64(v, o);
    v = (w < v) ? w : v;
  }
  return v;
}

__device__ __forceinline__ float wave_fsum(float v) {
#pragma unroll
  for (int o = 16; o > 0; o >>= 1) v += __shfl_xor(v, o, 32);
  return v;
}

// ---- block (256-thread, 8-wave) reductions --------------------------------
__device__ __forceinline__ unsigned long long
block_umax(unsigned long long v, unsigned long long* red, int wid, int lane) {
  v = wave_umax(v);
  __syncthreads();                 // protect red[] from previous round's readers
  if (lane == 0) red[wid] = v;
  __syncthreads();
  unsigned long long m = red[0];
#pragma unroll
  for (int w = 1; w < WAVES; ++w) m = (red[w] > m) ? red[w] : m;
  return m;
}

__device__ __forceinline__ unsigned long long
block_umin(unsigned long long v, unsigned long long* red, int wid, int lane) {
  v = wave_umin(v);
  __syncthreads();
  if (lane == 0) red[wid] = v;
  __syncthreads();
  unsigned long long m = red[0];
#pragma unroll
  for (int w = 1; w < WAVES; ++w) m = (red[w] < m) ? red[w] : m;
  return m;
}

__device__ __forceinline__ float
block_fsum(float v, float* red, int wid, int lane) {
  v = wave_fsum(v);
  __syncthreads();
  if (lane == 0) red[wid] = v;
  __syncthreads();
  float s = 0.f;
#pragma unroll
  for (int w = 0; w < WAVES; ++w) s += red[w];
  return s;
}

__device__ __forceinline__ float gelu_f(float x) {
  const float c = 0.7978845608028654f;  // sqrt(2/pi)
  float t = tanhf(c * (x + 0.044715f * x * x * x));
  return 0.5f * x * (1.0f + t);
}

// local candidate scans (key encodes value + index tie-break)
__device__ __forceinline__ unsigned long long
local_top_key(unsigned mask, const float* z, int tid) {
  unsigned long long best = 0ull;
#pragma unroll
  for (int i = 0; i < NSLOT; ++i) {
    if (!((mask >> i) & 1u)) {
      unsigned ab = __float_as_uint(z[i]) & 0x7fffffffu;     // |z| bits, monotonic
      unsigned j  = (unsigned)(tid + (i << 8));
      unsigned long long k = ((unsigned long long)ab << 32) | (0xffffffffu - j);
      best = (k > best) ? k : best;                          // ties -> smaller j
    }
  }
  return best;
}

__device__ __forceinline__ unsigned long long
local_bot_key(unsigned mask, const float* z, int tid) {
  unsigned long long best = ~0ull;
#pragma unroll
  for (int i = 0; i < NSLOT; ++i) {
    if (!((mask >> i) & 1u)) {
      unsigned ab = __float_as_uint(z[i]) & 0x7fffffffu;
      unsigned j  = (unsigned)(tid + (i << 8));
      unsigned long long k = ((unsigned long long)ab << 32) | j;  // ties -> smaller j
      best = (k < best) ? k : best;
    }
  }
  return best;
}

__global__ void __launch_bounds__(BLK)
fused_gelu_topk_gate(const float* __restrict__ x,
                     const float* __restrict__ ema_mean,
                     const float* __restrict__ ema_sq,
                     const float* __restrict__ ema_out,
                     const float* __restrict__ p_log_tau,
                     const float* __restrict__ p_log_beta_up,
                     const float* __restrict__ p_log_gamma,
                     const float* __restrict__ p_logit_beta_fam,
                     float* __restrict__ out) {
  __shared__ float xs[DD];                        // 16 KB staged row
  __shared__ unsigned long long redU[WAVES];
  __shared__ float redF[WAVES];

  const int tid  = threadIdx.x;
  const int lane = tid & 31;
  const int wid  = tid >> 5;
  const size_t row = blockIdx.x;
  const float* xrow = x + row * (size_t)DD;

  // -- Stage A: kick off async copy of the row into LDS (256 thr * 4 * 16B) --
#pragma unroll
  for (int c = 0; c < 4; ++c) {
    int off = (c * BLK + tid) * 16;  // byte offset within row
    async_b128_to_lds((const float*)((const char*)xrow + off),
                      (float*)((char*)xs + off));
  }

  // -- Stage B (overlapped with the copy): scalars + per-column stats --------
  float tau, beta_up, gamma, beta_fam;
  {
    float lt = p_log_tau[0];
    float lb = p_log_beta_up[0];
    float lg = p_log_gamma[0];
    float lf = p_logit_beta_fam[0];
    tau      = expf(lt);
    beta_up  = (lb > 20.f) ? lb : log1pf(expf(lb));   // softplus
    gamma    = (lg > 20.f) ? lg : log1pf(expf(lg));   // softplus
    beta_fam = 1.0f / (1.0f + expf(-lf));             // sigmoid
  }

  float mu[NSLOT], inv[NSLOT], eo[NSLOT];
  float s_ee = 0.f;
#pragma unroll
  for (int i = 0; i < NSLOT; ++i) {
    int j = tid + (i << 8);
    float m = ema_mean[j];
    float v = fmaxf(ema_sq[j] - m * m, 1e-4f);
    mu[i]  = m;
    inv[i] = 1.0f / (sqrtf(v) + 1e-5f);
    float e = ema_out[j];
    eo[i]  = e;
    s_ee  += e * e;
  }

  wait_async0();      // this wave's async copies landed in LDS
  __syncthreads();    // all 8 waves done

  // -- Stage C: gelu, z-score, row dot-products ------------------------------
  float z[NSLOT], g[NSLOT];
  float s_gg = 0.f, s_ge = 0.f;
#pragma unroll
  for (int i = 0; i < NSLOT; ++i) {
    int j = tid + (i << 8);
    float xv = xs[j];
    float gv = gelu_f(xv);
    float zv = (xv - mu[i]) * inv[i];
    g[i] = gv;
    z[i] = zv;
    s_gg += gv * gv;
    s_ge += gv * eo[i];
  }

  float n_gg = block_fsum(s_gg, redF, wid, lane);
  float n_ge = block_fsum(s_ge, redF, wid, lane);
  float n_ee = block_fsum(s_ee, redF, wid, lane);
  float cs = n_ge / (fmaxf(sqrtf(n_gg), 1e-12f) * fmaxf(sqrtf(n_ee), 1e-12f));
  cs = fminf(fmaxf(cs, -1.0f), 1.0f);
  const float gcos = expf(-tau * cs);

  // -- Stage D: iterative block arg-max/min, k=16 each (ties -> lowest idx) --
  unsigned topm = 0u, botm = 0u;
  unsigned long long loc = local_top_key(topm, z, tid);
  for (int it = 0; it < KSEL; ++it) {
    unsigned long long best = block_umax(loc, redU, wid, lane);
    unsigned jwin = 0xffffffffu - (unsigned)best;
    if ((jwin & 255u) == (unsigned)tid) {       // winner owns this column
      topm |= 1u << (jwin >> 8);
      loc = local_top_key(topm, z, tid);        // only winner rescans
    }
  }
  loc = local_bot_key(botm, z, tid);
  for (int it = 0; it < KSEL; ++it) {
    unsigned long long best = block_umin(loc, redU, wid, lane);
    unsigned jwin = (unsigned)best;
    if ((jwin & 255u) == (unsigned)tid) {
      botm |= 1u << (jwin >> 8);
      loc = local_bot_key(botm, z, tid);
    }
  }

  // -- Stage E: apply gates (bot-k overwrites top-k, as in reference) --------
  float* orow = out + row * (size_t)DD;
#pragma unroll
  for (int i = 0; i < NSLOT; ++i) {
    int j = tid + (i << 8);
    float gate = 1.0f;
    if ((botm >> i) & 1u) {
      gate = beta_fam;
    } else if ((topm >> i) & 1u) {
      float gt = 1.0f + beta_up * tanhf(gamma * z[i]);
      gate = fminf(fmaxf(gt, 0.1f), 8.0f);
    }
    orow[j] = g[i] * gate * gcos;
  }
}

extern "C" void kernel_launch(void* const* d_in, const int* in_sizes, int n_in,
                              void* d_out, int out_size, void* d_ws, size_t ws_size,
                              hipStream_t stream) {
  (void)n_in; (void)out_size; (void)d_ws; (void)ws_size;
  const float* x        = (const float*)d_in[0];
  const float* ema_mean = (const float*)d_in[1];
  const float* ema_sq   = (const float*)d_in[2];
  const float* ema_out  = (const float*)d_in[3];
  const float* ltau     = (const float*)d_in[4];
  const float* lbup     = (const float*)d_in[5];
  const float* lgam     = (const float*)d_in[6];
  const float* lbfam    = (const float*)d_in[7];
  float* out = (float*)d_out;

  const int rows = in_sizes[0] / DD;  // B*T = 16384
  fused_gelu_topk_gate<<<dim3(rows), dim3(BLK), 0, stream>>>(
      x, ema_mean, ema_sq, ema_out, ltau, lbup, lgam, lbfam, out);
}